// TTStack_85787676770424
// MI455X (gfx1250) — compile-verified
//
#include <hip/hip_runtime.h>

// TT-stack contraction: out[b, 512, 512] from 3 gathered TT cores.
// 1 workgroup (256 thr = 8 wave32) per batch element.
//   Stage 1: t[(l0',r0),(l1,r1,k2)] = c0 x c1   (16x16x1024 matmul, WMMA f32 16x16x4)
//   Stage 2: out = t' x c2'                     (1024x16x64 matmul, WMMA f32 16x16x4)
// l0 processed in pairs so stage-1 M fills the 16-row WMMA tile.
// Store-bound kernel (512MB f32 out @ 23.3TB/s ~= 23us floor); WMMA f32 keeps the
// ~5.4 GFLOP of contraction math off the critical path without precision loss.

typedef __attribute__((ext_vector_type(2))) float v2f;
typedef __attribute__((ext_vector_type(8))) float v8f;

__device__ __forceinline__ v8f wmma_f32_k4(v2f a, v2f b, v8f c) {
    // D = A(16x4 f32) * B(4x16 f32) + C(16x16 f32)
    return __builtin_amdgcn_wmma_f32_16x16x4_f32(
        /*neg_a=*/false, a, /*neg_b=*/false, b,
        /*c_mod=*/(short)0, c, /*reuse_a=*/false, /*reuse_b=*/false);
}

__global__ __launch_bounds__(256, 4)   // 4 waves/SIMD: 2 WGs per 320KB WGP (2x136KB LDS)
void tt_stack_wmma_kernel(const float* __restrict__ core0,   // [STACK,8,8,16]
                          const float* __restrict__ core1,   // [STACK,16,8,8,16]
                          const float* __restrict__ core2,   // [STACK,16,8,8]
                          const int*   __restrict__ indices, // [B]
                          float*       __restrict__ out)     // [B,512,512]
{
    __shared__ float lds_c1[16 * 1024];  // 64KB  [k1][l1*128 + r1*16 + k2]
    __shared__ float lds_t [16 * 1024];  // 64KB  [(l0p*8 + r0)][l1*128 + r1*16 + k2]
    __shared__ float lds_c0[1024];       //  4KB  [l0*128 + r0*16 + k1]
    __shared__ float lds_c2[1024];       //  4KB  [k2*64 + l2*8 + r2]

    const int b    = blockIdx.x;
    const int tid  = threadIdx.x;        // 0..255
    const int wave = tid >> 5;           // 0..7
    const int lane = tid & 31;
    const int lo   = lane & 15;          // lane within half-wave (M/N index)
    const int hi   = lane >> 4;          // half-wave select (K pairing per ISA layout)

    const int s = indices[b];

    // ---- cooperative global -> LDS loads (float4) ----
    {
        const float4* g1 = (const float4*)(core1 + (size_t)s * 16384);
        float4* l1 = (float4*)lds_c1;
        #pragma unroll
        for (int i = 0; i < 16; ++i) l1[tid + 256 * i] = g1[tid + 256 * i];
        ((float4*)lds_c0)[tid] = ((const float4*)(core0 + (size_t)s * 1024))[tid];
        ((float4*)lds_c2)[tid] = ((const float4*)(core2 + (size_t)s * 1024))[tid];
    }
    __syncthreads();

    // ---- preload stage-2 B fragments from c2 (constant across all l0 pairs) ----
    // B tile (4x16): lanes0-15 -> K = kk*4 + {0,1}, lanes16-31 -> K = kk*4 + {2,3}; N = lo
    v2f bf2[4][4];
    #pragma unroll
    for (int kk = 0; kk < 4; ++kk) {
        #pragma unroll
        for (int nt = 0; nt < 4; ++nt) {
            const int k = kk * 4 + hi * 2;
            const int n = nt * 16 + lo;
            bf2[kk][nt].x = lds_c2[k * 64 + n];
            bf2[kk][nt].y = lds_c2[(k + 1) * 64 + n];
        }
    }

    float* const outb = out + (size_t)b * 512 * 512;
    const v8f vzero = {0.f, 0.f, 0.f, 0.f, 0.f, 0.f, 0.f, 0.f};

    #pragma unroll 1
    for (int p = 0; p < 4; ++p) {        // l0 pair: l0 = p*2 + l0'
        // ======== Stage 1: t[(l0',r0)][(l1,r1,k2)] = c0_pair (16x16) x c1 (16x1024)
        // A tile rows M = (l0'*8 + r0) = lo ; K = k1
        v2f af1[4];
        {
            const int l0 = p * 2 + (lo >> 3);
            const int r0 = lo & 7;
            #pragma unroll
            for (int kk = 0; kk < 4; ++kk) {
                const int k1 = kk * 4 + hi * 2;
                af1[kk] = *(const v2f*)(lds_c0 + l0 * 128 + r0 * 16 + k1); // 8B aligned
            }
        }
        #pragma unroll 4
        for (int jj = 0; jj < 8; ++jj) {             // this wave's N-tiles
            const int j16 = wave * 8 + jj;
            const int n   = j16 * 16 + lo;
            v8f acc = vzero;
            #pragma unroll
            for (int kk = 0; kk < 4; ++kk) {
                const int k1 = kk * 4 + hi * 2;
                v2f bf;
                bf.x = lds_c1[k1 * 1024 + n];
                bf.y = lds_c1[(k1 + 1) * 1024 + n];
                acc = wmma_f32_k4(af1[kk], bf, acc);
            }
            // D tile: VGPR r holds M = r + 8*hi, N = lo
            float* tp = lds_t + (8 * hi) * 1024 + n;
            #pragma unroll
            for (int r = 0; r < 8; ++r) tp[r * 1024] = acc[r];
        }
        __syncthreads();

        // ======== Stage 2: out_rows(pair) = t' (1024x16) x c2' (16x64)
        // m = l0'*512 + l1*64 + r0*8 + r1 ; n = l2*8 + r2
        #pragma unroll 2
        for (int mi = 0; mi < 8; ++mi) {             // this wave's M-tiles
            const int mt = wave * 8 + mi;
            // A fragment: row m = mt*16 + lo, K = k2 (contiguous in lds_t)
            const int m    = mt * 16 + lo;
            const int trow = ((m >> 9) << 3) | ((m >> 3) & 7);        // l0'*8 + r0
            const int tcol = (((m >> 6) & 7) << 7) | ((m & 7) << 4);  // l1*128 + r1*16
            const float* tptr = lds_t + trow * 1024 + tcol;
            v2f af[4];
            #pragma unroll
            for (int kk = 0; kk < 4; ++kk)
                af[kk] = *(const v2f*)(tptr + kk * 4 + hi * 2);       // 8B aligned

            // Store decomposition: across the 8 acc regs only r1 (=r) varies.
            const int mh   = mt * 2 + hi;            // = M' >> 3 for all 8 rows
            const int l0p  = mh >> 6;
            const int l1b  = (mh >> 3) & 7;
            const int r0b  = mh & 7;
            const int rowb = (p * 2 + l0p) * 64 + l1b * 8;  // + l2
            const int colb = r0b * 64;                      // + r*8 + r2

            #pragma unroll
            for (int nt = 0; nt < 4; ++nt) {
                v8f acc = vzero;
                #pragma unroll
                for (int kk = 0; kk < 4; ++kk)
                    acc = wmma_f32_k4(af[kk], bf2[kk][nt], acc);
                const int n   = nt * 16 + lo;
                const int l2  = n >> 3;
                const int r2  = n & 7;
                const int off = ((rowb + l2) << 9) + colb + r2;   // 32-bit, < 2^18
                float* op = outb + off;
                #pragma unroll
                for (int r = 0; r < 8; ++r) op[r * 8] = acc[r];
            }
        }
        __syncthreads();
    }
}

extern "C" void kernel_launch(void* const* d_in, const int* in_sizes, int n_in,
                              void* d_out, int out_size, void* d_ws, size_t ws_size,
                              hipStream_t stream) {
    (void)n_in; (void)out_size; (void)d_ws; (void)ws_size;
    const float* core0   = (const float*)d_in[0];
    const float* core1   = (const float*)d_in[1];
    const float* core2   = (const float*)d_in[2];
    const int*   indices = (const int*)d_in[3];
    float*       out     = (float*)d_out;
    const int batch = in_sizes[3];   // 512
    dim3 grid(batch), block(256);
    tt_stack_wmma_kernel<<<grid, block, 0, stream>>>(core0, core1, core2, indices, out);
}